// HMLDM_37065567765088
// MI455X (gfx1250) — compile-verified
//
#include <hip/hip_runtime.h>

typedef __attribute__((ext_vector_type(2))) float v2f;
typedef __attribute__((ext_vector_type(8))) float v8f;

#define NROWS 16384
#define MCOLS 8192
#define DDIM  8
#define EPS_F 1e-6f

#define TILE_BLOCKS 4096        // x 8 waves = 32768 waves
#define WAVES_PER_STRIP 32      // waves sharing one 16-row strip
#define TJ_PER_WAVE ((MCOLS / 16) / WAVES_PER_STRIP)  // 512/32 = 16
#define EDGE_BLOCKS 2048

__device__ __forceinline__ float fast_sqrtf(float x) {
  return __builtin_amdgcn_sqrtf(x);   // raw v_sqrt_f32 (~1 ulp)
}

// ---------------------------------------------------------------------------
// Kernel 1: softmax over D=8 per row + row sum-of-squares, for both z and w.
// ---------------------------------------------------------------------------
__global__ void hm_softmax_rows(const float* __restrict__ lz,
                                const float* __restrict__ lw,
                                float* __restrict__ zbuf,
                                float* __restrict__ wbuf,
                                float* __restrict__ zz,
                                float* __restrict__ ww) {
  int t = blockIdx.x * blockDim.x + threadIdx.x;
  if (t >= NROWS + MCOLS) return;
  const float* src;
  float* dst;
  float* ss;
  if (t < NROWS) {
    src = lz + (size_t)t * DDIM;  dst = zbuf + (size_t)t * DDIM;  ss = zz + t;
  } else {
    int r = t - NROWS;
    src = lw + (size_t)r * DDIM;  dst = wbuf + (size_t)r * DDIM;  ss = ww + r;
  }
  float v[DDIM];
  float mx = -3.402823466e38f;
#pragma unroll
  for (int k = 0; k < DDIM; ++k) { v[k] = src[k]; mx = fmaxf(mx, v[k]); }
  float s = 0.f;
#pragma unroll
  for (int k = 0; k < DDIM; ++k) { v[k] = __expf(v[k] - mx); s += v[k]; }
  float inv = 1.f / s;
  float sq = 0.f;
#pragma unroll
  for (int k = 0; k < DDIM; ++k) { v[k] *= inv; sq += v[k] * v[k]; dst[k] = v[k]; }
  *ss = sq;
}

// ---------------------------------------------------------------------------
// Kernel 2: pairwise term. One wave32 per 16x16 tile; each wave owns one
// 16-row strip (ti fixed) and sweeps 16 consecutive tj tiles so the A operand
// (z rows) and row-side epilogue data (zz, gamma_rows) hoist out of the loop.
// dot(z_i, w_j) via two V_WMMA_F32_16X16X4_F32 (K=8 split into 2x K=4).
// A (16x4 f32): VGPR0 = K0 (lanes 0-15) / K2 (lanes 16-31), VGPR1 = K1/K3.
// B (4x16 f32): same lane/K split, lane index = column j.
// C/D (16x16 f32, 8 VGPRs): VGPR r -> row r (+8 for lanes 16-31), col = lane&15.
// Epilogue fuses fast sqrt + single exp(gr+gc-(dist+eps)) per element.
// ---------------------------------------------------------------------------
__global__ void __launch_bounds__(256)
hm_tile_kernel(const float* __restrict__ zbuf,
               const float* __restrict__ wbuf,
               const float* __restrict__ zz,
               const float* __restrict__ ww,
               const float* __restrict__ gr,
               const float* __restrict__ gc,
               float* __restrict__ partials) {
  const int lane   = threadIdx.x & 31;
  const int wave   = threadIdx.x >> 5;
  const int laneLo = lane & 15;
  const int laneHi = lane >> 4;   // 0: K/row group 0, 1: K/row group 1

  // 32768 global waves = 1024 strips x 32 waves; each wave does 16 tj tiles.
  const int gwave = blockIdx.x * 8 + wave;
  const int ti    = gwave / WAVES_PER_STRIP;          // 0..1023
  const int tj0   = (gwave % WAVES_PER_STRIP) * TJ_PER_WAVE;
  const int i0    = ti * 16;

  // ---- hoisted A operand (z rows of this strip) ----
  const float* za = zbuf + (size_t)(i0 + laneLo) * DDIM + laneHi * 2;
  v2f aLo; aLo[0] = za[0]; aLo[1] = za[1];
  v2f aHi; aHi[0] = za[4]; aHi[1] = za[5];

  // ---- hoisted row-side epilogue data ----
  const int ibase = i0 + laneHi * 8;
  const float4 zzA = *(const float4*)(zz + ibase);
  const float4 zzB = *(const float4*)(zz + ibase + 4);
  const float4 grA = *(const float4*)(gr + ibase);
  const float4 grB = *(const float4*)(gr + ibase + 4);
  const float zzv[8] = {zzA.x, zzA.y, zzA.z, zzA.w, zzB.x, zzB.y, zzB.z, zzB.w};
  const float grv[8] = {grA.x, grA.y, grA.z, grA.w, grB.x, grB.y, grB.z, grB.w};

  float sum = 0.f;

#pragma unroll 1
  for (int tj = tj0; tj < tj0 + TJ_PER_WAVE; ++tj) {
    const int j0 = tj * 16;

    // B operand: rows of w (columns of the tile)
    const float* wb = wbuf + (size_t)(j0 + laneLo) * DDIM + laneHi * 2;
    v2f bLo; bLo[0] = wb[0]; bLo[1] = wb[1];
    v2f bHi; bHi[0] = wb[4]; bHi[1] = wb[5];

    v8f acc = {};
    acc = __builtin_amdgcn_wmma_f32_16x16x4_f32(
        false, aLo, false, bLo, (short)0, acc, false, false);
    acc = __builtin_amdgcn_wmma_f32_16x16x4_f32(
        false, aHi, false, bHi, (short)0, acc, false, false);

    const int j     = j0 + laneLo;
    const float wwj = ww[j];
    const float gcj = gc[j];

#pragma unroll
    for (int r = 0; r < 8; ++r) {
      float sq   = zzv[r] + wwj - 2.f * acc[r];
      float dist = fast_sqrtf(fmaxf(sq, 0.f));
      sum += __expf(grv[r] + gcj - (dist + EPS_F));
    }
  }

  // Wave reduction (wave32), then per-block LDS reduction.
#pragma unroll
  for (int o = 16; o > 0; o >>= 1) sum += __shfl_down(sum, o, 32);
  __shared__ float ls[8];
  if (lane == 0) ls[wave] = sum;
  __syncthreads();
  if (threadIdx.x == 0) {
    float b = 0.f;
#pragma unroll
    for (int i = 0; i < 8; ++i) b += ls[i];
    partials[blockIdx.x] = b;
  }
}

// ---------------------------------------------------------------------------
// Kernel 3: edge term. Grid-stride over E edges; per-block partial sums.
// ---------------------------------------------------------------------------
__global__ void hm_edge_kernel(const float* __restrict__ zbuf,
                               const float* __restrict__ wbuf,
                               const float* __restrict__ gr,
                               const float* __restrict__ gc,
                               const float* __restrict__ wts,
                               const int* __restrict__ ri,
                               const int* __restrict__ ci,
                               int E,
                               float* __restrict__ partials) {
  float sum = 0.f;
  for (int e = blockIdx.x * blockDim.x + threadIdx.x; e < E;
       e += gridDim.x * blockDim.x) {
    const int r = ri[e];
    const int c = ci[e];
    const float4 z0 = *(const float4*)(zbuf + (size_t)r * DDIM);
    const float4 z1 = *(const float4*)(zbuf + (size_t)r * DDIM + 4);
    const float4 w0 = *(const float4*)(wbuf + (size_t)c * DDIM);
    const float4 w1 = *(const float4*)(wbuf + (size_t)c * DDIM + 4);
    float d = 0.f;
    float dx;
    dx = z0.x - w0.x; d += dx * dx;
    dx = z0.y - w0.y; d += dx * dx;
    dx = z0.z - w0.z; d += dx * dx;
    dx = z0.w - w0.w; d += dx * dx;
    dx = z1.x - w1.x; d += dx * dx;
    dx = z1.y - w1.y; d += dx * dx;
    dx = z1.z - w1.z; d += dx * dx;
    dx = z1.w - w1.w; d += dx * dx;
    const float dist = fast_sqrtf(d);
    sum += wts[e] * (gr[r] + gc[c] - dist);
  }
  const int lane = threadIdx.x & 31;
  const int wave = threadIdx.x >> 5;
#pragma unroll
  for (int o = 16; o > 0; o >>= 1) sum += __shfl_down(sum, o, 32);
  __shared__ float ls[8];
  if (lane == 0) ls[wave] = sum;
  __syncthreads();
  if (threadIdx.x == 0) {
    float b = 0.f;
#pragma unroll
    for (int i = 0; i < 8; ++i) b += ls[i];
    partials[blockIdx.x] = b;
  }
}

// ---------------------------------------------------------------------------
// Kernel 4: deterministic final combine. out = z_pdist1 - z_pdist2.
// ---------------------------------------------------------------------------
__global__ void hm_final_kernel(const float* __restrict__ tp, int ntp,
                                const float* __restrict__ ep, int nep,
                                float* __restrict__ out) {
  __shared__ float s1s[256];
  __shared__ float s2s[256];
  float s1 = 0.f, s2 = 0.f;
  for (int i = threadIdx.x; i < ntp; i += blockDim.x) s1 += tp[i];
  for (int i = threadIdx.x; i < nep; i += blockDim.x) s2 += ep[i];
  s1s[threadIdx.x] = s1;
  s2s[threadIdx.x] = s2;
  __syncthreads();
  for (int o = blockDim.x >> 1; o > 0; o >>= 1) {
    if (threadIdx.x < (unsigned)o) {
      s1s[threadIdx.x] += s1s[threadIdx.x + o];
      s2s[threadIdx.x] += s2s[threadIdx.x + o];
    }
    __syncthreads();
  }
  if (threadIdx.x == 0) out[0] = s1s[0] - s2s[0];
}

// ---------------------------------------------------------------------------
extern "C" void kernel_launch(void* const* d_in, const int* in_sizes, int n_in,
                              void* d_out, int out_size, void* d_ws, size_t ws_size,
                              hipStream_t stream) {
  const float* gamma_rows = (const float*)d_in[0];  // [N]
  const float* gamma_cols = (const float*)d_in[1];  // [M]
  const float* latent_z   = (const float*)d_in[2];  // [N, 8]
  const float* latent_w   = (const float*)d_in[3];  // [M, 8]
  const float* weights    = (const float*)d_in[4];  // [E]
  const int*   rows_idx   = (const int*)d_in[5];    // [E]
  const int*   col_idx    = (const int*)d_in[6];    // [E]
  const int E = in_sizes[4];

  // Workspace layout (floats)
  float* ws   = (float*)d_ws;
  float* zbuf = ws;                        // N*8
  float* wbuf = zbuf + (size_t)NROWS * 8;  // M*8
  float* zz   = wbuf + (size_t)MCOLS * 8;  // N
  float* wwb  = zz + NROWS;                // M
  float* tp   = wwb + MCOLS;               // TILE_BLOCKS
  float* ep   = tp + TILE_BLOCKS;          // EDGE_BLOCKS

  // 1) softmax + sum-of-squares
  {
    int total = NROWS + MCOLS;
    int blk = 256;
    hm_softmax_rows<<<(total + blk - 1) / blk, blk, 0, stream>>>(
        latent_z, latent_w, zbuf, wbuf, zz, wwb);
  }

  // 2) pairwise WMMA tile kernel (fixed schedule: 32768 waves, 16 tiles each)
  hm_tile_kernel<<<TILE_BLOCKS, 256, 0, stream>>>(
      zbuf, wbuf, zz, wwb, gamma_rows, gamma_cols, tp);

  // 3) edge term
  hm_edge_kernel<<<EDGE_BLOCKS, 256, 0, stream>>>(
      zbuf, wbuf, gamma_rows, gamma_cols, weights, rows_idx, col_idx, E, ep);

  // 4) deterministic combine
  hm_final_kernel<<<1, 256, 0, stream>>>(tp, TILE_BLOCKS, ep, EDGE_BLOCKS,
                                         (float*)d_out);
}